// Dual_AMN_NR_GraphAttention_Layer_38963943309351
// MI455X (gfx1250) — compile-verified
//
#include <hip/hip_runtime.h>

#define NODE_SIZE 50000
#define REL_SIZE  500
#define HID       128
#define NTRIPLE   800000
#define EPSF      1e-12f

typedef __attribute__((ext_vector_type(2))) float v2f;
typedef __attribute__((ext_vector_type(8))) float v8f;

// ---------------------------------------------------------------------------
// Zero-fill (out and row_sums must be zeroed every call: graph replays)
// ---------------------------------------------------------------------------
__global__ void zero_kernel(float* __restrict__ p, int n) {
    int i = blockIdx.x * blockDim.x + threadIdx.x;
    int stride = gridDim.x * blockDim.x;
    for (; i < n; i += stride) p[i] = 0.0f;
}

// ---------------------------------------------------------------------------
// Per-relation stats. One wave (32 threads) per 16-row tile.
//   rel_att[r]  = rel_emb[r] . attn          (via V_WMMA_F32_16X16X4_F32)
//   rel_norm[r] = ||rel_emb[r]||             (wave reduction)
//
// A (16x4 f32): lanes 0-15 hold K={0,1}, lanes 16-31 hold K={2,3} of row M=lane&15.
// B (4x16 f32): v.x = row K=koff across lanes (N=lane&15), v.y = row K=koff+1;
//               only column N==0 carries the attn vector, rest zero.
// C (16x16 f32): VGPR j -> M=j (lanes 0-15) / M=j+8 (lanes 16-31), N=lane&15.
// ---------------------------------------------------------------------------
__global__ __launch_bounds__(32)
void rel_stats_kernel(const float* __restrict__ rel_emb,
                      const float* __restrict__ attn,
                      float* __restrict__ rel_norm,
                      float* __restrict__ rel_att) {
    const int lane = threadIdx.x & 31;
    const int tile = blockIdx.x;          // 32 tiles * 16 rows = 512 >= 500
    const int half = lane >> 4;           // 0 or 1
    const int m16  = lane & 15;

    const int   row   = tile * 16 + m16;
    const bool  rowok = row < REL_SIZE;
    const float amask = rowok ? 1.0f : 0.0f;
    // clamp pointer so OOB lanes still issue a valid load (keeps EXEC uniform)
    const float* arow = rel_emb + (size_t)(rowok ? row : 0) * HID;
    const float bmask = (m16 == 0) ? 1.0f : 0.0f;

    v8f c = {};
    for (int k = 0; k < HID; k += 4) {
        const int koff = half * 2;
        v2f a, b;
        a.x = arow[k + koff + 0] * amask;
        a.y = arow[k + koff + 1] * amask;
        b.x = attn[k + koff + 0] * bmask;
        b.y = attn[k + koff + 1] * bmask;
        c = __builtin_amdgcn_wmma_f32_16x16x4_f32(false, a, false, b,
                                                  (short)0, c, false, false);
    }
    if (m16 == 0) {
        const int mbase = tile * 16 + half * 8;
#pragma unroll
        for (int j = 0; j < 8; ++j)
            if (mbase + j < REL_SIZE) rel_att[mbase + j] = c[j];
    }

    // rel_norm: each lane sums squares over half a row (64 elems), combine halves
    float ss = 0.0f;
    {
        const float4* p = (const float4*)(arow + half * 64);
#pragma unroll
        for (int i = 0; i < 16; ++i) {
            float4 f = p[i];
            ss += f.x * f.x + f.y * f.y + f.z * f.z + f.w * f.w;
        }
    }
    ss += __shfl_xor(ss, 16, 32);
    if (half == 0 && rowok) rel_norm[row] = sqrtf(ss);
}

// ---------------------------------------------------------------------------
// Pass 1: att -> exp -> atomic row sums.  One thread per triple.
// ---------------------------------------------------------------------------
__global__ __launch_bounds__(256)
void att_kernel(const int*   __restrict__ indices,
                const int*   __restrict__ sp_idx,
                const float* __restrict__ sp_val,
                const float* __restrict__ rel_norm,
                const float* __restrict__ rel_att,
                float* __restrict__ exp_v,
                float* __restrict__ row_sums) {
    int t = blockIdx.x * blockDim.x + threadIdx.x;
    if (t >= NTRIPLE) return;
    const int   r    = sp_idx[2 * t + 1];
    const float v    = sp_val[t];
    const float coef = v / fmaxf(fabsf(v) * rel_norm[r], EPSF);
    const float e    = expf(coef * rel_att[r]);
    exp_v[t] = e;
    atomicAdd(&row_sums[indices[2 * t]], e);
}

// ---------------------------------------------------------------------------
// Pass 2: Householder reflection + softmax-weighted scatter.
// One wave32 per triple; each lane owns 4 consecutive floats (float4).
// Feature/relation rows are single coalesced 512B transactions (hot in L2).
// ---------------------------------------------------------------------------
__global__ __launch_bounds__(256)
void scatter_kernel(const float* __restrict__ features,
                    const float* __restrict__ rel_emb,
                    const int*   __restrict__ indices,
                    const int*   __restrict__ sp_idx,
                    const float* __restrict__ sp_val,
                    const float* __restrict__ rel_norm,
                    const float* __restrict__ exp_v,
                    const float* __restrict__ row_sums,
                    float* __restrict__ out) {
    const int wavesPerBlock = blockDim.x >> 5;
    const int t    = blockIdx.x * wavesPerBlock + (threadIdx.x >> 5);
    const int lane = threadIdx.x & 31;
    if (t >= NTRIPLE) return;

    const int   row  = indices[2 * t + 0];
    const int   col  = indices[2 * t + 1];
    const int   r    = sp_idx[2 * t + 1];
    const float v    = sp_val[t];
    const float coef = v / fmaxf(fabsf(v) * rel_norm[r], EPSF);

    const float4 f4 = ((const float4*)(features + (size_t)col * HID))[lane];
    const float4 r4 = ((const float4*)(rel_emb  + (size_t)r   * HID))[lane];

    // dot(features[col], rel_emb[r]) across the wave
    float partial = f4.x * r4.x + f4.y * r4.y + f4.z * r4.z + f4.w * r4.w;
#pragma unroll
    for (int off = 16; off >= 1; off >>= 1)
        partial += __shfl_xor(partial, off, 32);

    const float proj = coef * partial;          // neighs . rels_sum_unit
    const float s    = exp_v[t] / row_sums[row];
    const float k2   = 2.0f * proj * coef;      // folds unit-vector scale back in

    float* dst = out + (size_t)row * HID + lane * 4;
    atomicAdd(dst + 0, (f4.x - k2 * r4.x) * s);
    atomicAdd(dst + 1, (f4.y - k2 * r4.y) * s);
    atomicAdd(dst + 2, (f4.z - k2 * r4.z) * s);
    atomicAdd(dst + 3, (f4.w - k2 * r4.w) * s);
}

// ---------------------------------------------------------------------------
extern "C" void kernel_launch(void* const* d_in, const int* in_sizes, int n_in,
                              void* d_out, int out_size, void* d_ws, size_t ws_size,
                              hipStream_t stream) {
    const float* features = (const float*)d_in[0];
    const float* rel_emb  = (const float*)d_in[1];
    const int*   indices  = (const int*)  d_in[2];
    const int*   sp_idx   = (const int*)  d_in[3];
    const float* sp_val   = (const float*)d_in[4];
    const float* attn     = (const float*)d_in[5];
    float* out = (float*)d_out;

    float* ws       = (float*)d_ws;
    float* rel_norm = ws;                      // 512 floats
    float* rel_att  = ws + 512;                // 512 floats
    float* exp_v    = ws + 1024;               // NTRIPLE floats
    float* row_sums = ws + 1024 + NTRIPLE;     // NODE_SIZE floats

    // zero output accumulator and row sums (every call — graph replays)
    zero_kernel<<<2048, 256, 0, stream>>>(out, NODE_SIZE * HID);
    zero_kernel<<<256, 256, 0, stream>>>(row_sums, NODE_SIZE);

    // per-relation stats (WMMA matvec + norms): 512 padded rows / 16 = 32 tiles
    rel_stats_kernel<<<32, 32, 0, stream>>>(rel_emb, attn, rel_norm, rel_att);

    // pass 1: exp(att) + segment row sums
    att_kernel<<<(NTRIPLE + 255) / 256, 256, 0, stream>>>(
        indices, sp_idx, sp_val, rel_norm, rel_att, exp_v, row_sums);

    // pass 2: reflection + weighted scatter (8 waves per block, 1 wave/triple)
    scatter_kernel<<<(NTRIPLE + 7) / 8, 256, 0, stream>>>(
        features, rel_emb, indices, sp_idx, sp_val, rel_norm,
        exp_v, row_sums, out);
}